// AxisAttention_75376676044825
// MI455X (gfx1250) — compile-verified
//
#include <hip/hip_runtime.h>

// out[i,j,:] = g[i,j,:] + 384*(v[i,j,:] + v[j,i,:]),  v = g @ Wv + bv
// (softmax over a size-1 axis in the reference == 1.0, so q/k are dead)

typedef __attribute__((ext_vector_type(2))) float v2f;
typedef __attribute__((ext_vector_type(8))) float v8f;

#define GN     384   // grid side
#define KD     512   // feature dim (K and N of the GEMM)
#define TILE   8     // (i,j) tile side -> 64 pairs per block
#define BM     128   // v-rows per block: 64 direct + 64 transposed
#define BN     64    // output-column chunk
#define KT     32    // K slab staged through LDS
#define APITCH 36    // dwords per A row: 144 B (16-B aligned for b128)
#define BPPITCH 132  // dwords per B pair-row (kp): 128 data + 4 pad, 528 B
#define CPITCH 68    // dwords per C row: 272 B

__global__ __launch_bounds__(256)
void axis_attn_fused_wmma(const float* __restrict__ g,
                          const float* __restrict__ Wv,
                          const float* __restrict__ bv,
                          float* __restrict__ out) {
    // union: A/B K-slabs during GEMM, C tile during epilogue
    __shared__ __align__(16) float smem[BM * CPITCH];   // 34816 B
    float* As = smem;                   // [128][36]
    float* Bs = smem + BM * APITCH;     // [16][132] pair-interleaved: Bs[k/2][n][2]
    float* Cs = smem;                   // [128][68] (reused after barrier)

    const int tid  = threadIdx.x;
    const int wave = tid >> 5;
    const int lane = tid & 31;
    const int hf   = lane >> 4;   // lane half (ISA K split)
    const int l15  = lane & 15;
    const int wm   = wave >> 1;   // 0..3 -> 32-row strip
    const int wn   = wave & 1;    // 0..1 -> 32-col strip
    const int m_off = wm * 32;
    const int n_off = wn * 32;

    const int i0 = blockIdx.y * TILE;
    const int j0 = blockIdx.x * TILE;

    for (int nc = 0; nc < KD / BN; ++nc) {
        const int n0 = nc * BN;
        v8f acc00 = {}, acc01 = {}, acc10 = {}, acc11 = {};

        for (int kk = 0; kk < KD; kk += KT) {
            __syncthreads();   // previous slab consumed / Cs epilogue done

            // ---- stage A: 128 rows x 32 K, async memory->LDS (ASYNCcnt) ----
            // rows 0..63 = (i,j) pairs, rows 64..127 = (j,i) transposed pairs
            #pragma unroll
            for (int it = 0; it < 4; ++it) {
                int idx = tid + it * 256;
                int row = idx >> 3;
                int c   = (idx & 7) << 2;
                int rr  = row & 63;
                int hi  = rr >> 3, lo = rr & 7;
                int m   = (row < 64) ? ((i0 + hi) * GN + (j0 + lo))
                                     : ((j0 + hi) * GN + (i0 + lo));
                const float* gp = g + (size_t)m * KD + kk + c;
                // LDS byte address = low 32 bits of the flat shared pointer
                unsigned ldsb = (unsigned)(size_t)(As + row * APITCH + c);
                asm volatile("global_load_async_to_lds_b128 %0, %1, off"
                             :: "v"(ldsb), "v"(gp) : "memory");
                if (kk + KT < KD)   // warm L2 for next slab -> global_prefetch_b8
                    __builtin_prefetch(gp + KT, 0, 1);
            }

            // ---- stage B: 32 K x 64 cols of Wv, pair-interleaved in LDS ----
            {
                int kp = tid >> 4;          // 0..15 -> K rows {2kp, 2kp+1}
                int c  = (tid & 15) << 2;   // col group of 4
                const float* bp = Wv + (size_t)(kk + 2 * kp) * KD + n0 + c;
                float4 r0 = *(const float4*)(bp);
                float4 r1 = *(const float4*)(bp + KD);
                float* dst = Bs + kp * BPPITCH + c * 2;
                v2f t0; t0.x = r0.x; t0.y = r1.x; *(v2f*)(dst + 0) = t0;
                v2f t1; t1.x = r0.y; t1.y = r1.y; *(v2f*)(dst + 2) = t1;
                v2f t2; t2.x = r0.z; t2.y = r1.z; *(v2f*)(dst + 4) = t2;
                v2f t3; t3.x = r0.w; t3.y = r1.w; *(v2f*)(dst + 6) = t3;
            }

            asm volatile("s_wait_asynccnt 0x0" ::: "memory");  // A in LDS
            __syncthreads();                                    // all waves staged

            // ---- 8 x (K=4) WMMA steps per slab ----
            #pragma unroll
            for (int k0 = 0; k0 < KT; k0 += 4) {
                const int ka = k0 + 2 * hf;    // lanes 0-15: K{k0,k0+1}; 16-31: K{k0+2,k0+3}
                const int kp = ka >> 1;        // pair-row in Bs
                v2f a0 = *(const v2f*)(As + (m_off +      l15) * APITCH + ka);
                v2f a1 = *(const v2f*)(As + (m_off + 16 + l15) * APITCH + ka);
                v2f b0 = *(const v2f*)(Bs + kp * BPPITCH + (n_off +      l15) * 2);
                v2f b1 = *(const v2f*)(Bs + kp * BPPITCH + (n_off + 16 + l15) * 2);
                acc00 = __builtin_amdgcn_wmma_f32_16x16x4_f32(false, a0, false, b0, (short)0, acc00, false, false);
                acc01 = __builtin_amdgcn_wmma_f32_16x16x4_f32(false, a0, false, b1, (short)0, acc01, false, false);
                acc10 = __builtin_amdgcn_wmma_f32_16x16x4_f32(false, a1, false, b0, (short)0, acc10, false, false);
                acc11 = __builtin_amdgcn_wmma_f32_16x16x4_f32(false, a1, false, b1, (short)0, acc11, false, false);
            }
        }

        __syncthreads();   // A/B slab reads done; reuse smem as C tile
        // ---- dump C (ISA layout: VGPR r -> rows {r, r+8}, lane half selects) ----
        #pragma unroll
        for (int r = 0; r < 8; ++r) {
            int rowa = m_off + r + 8 * hf;
            Cs[(rowa     ) * CPITCH + n_off +      l15] = acc00[r];
            Cs[(rowa     ) * CPITCH + n_off + 16 + l15] = acc01[r];
            Cs[(rowa + 16) * CPITCH + n_off +      l15] = acc10[r];
            Cs[(rowa + 16) * CPITCH + n_off + 16 + l15] = acc11[r];
        }
        __syncthreads();

        // ---- epilogue: 64 pair-rows x 64 cols ----
        const int p  = tid >> 2;          // pair (ii,jj), row p = ii*8+jj
        const int q  = tid & 3;           // 16-col quarter
        const int ii = p >> 3, jj = p & 7;
        const int bRow = 64 + jj * 8 + ii;   // transposed partner v[j,i]
        const size_t m = (size_t)(i0 + ii) * GN + (j0 + jj);
        #pragma unroll
        for (int c4 = 0; c4 < 4; ++c4) {
            int col = q * 16 + c4 * 4;
            float4 Ca = *(float4*)(Cs + p    * CPITCH + col);
            float4 Cb = *(float4*)(Cs + bRow * CPITCH + col);
            float4 gg = *(const float4*)(g  + m * KD + n0 + col);
            float4 bb = *(const float4*)(bv + n0 + col);
            float4 o;
            o.x = gg.x + 384.0f * (Ca.x + Cb.x + 2.0f * bb.x);
            o.y = gg.y + 384.0f * (Ca.y + Cb.y + 2.0f * bb.y);
            o.z = gg.z + 384.0f * (Ca.z + Cb.z + 2.0f * bb.z);
            o.w = gg.w + 384.0f * (Ca.w + Cb.w + 2.0f * bb.w);
            *(float4*)(out + m * KD + n0 + col) = o;
        }
        // next nc's first __syncthreads() protects Cs-as-As reuse
    }
}

extern "C" void kernel_launch(void* const* d_in, const int* in_sizes, int n_in,
                              void* d_out, int out_size, void* d_ws, size_t ws_size,
                              hipStream_t stream) {
    (void)in_sizes; (void)n_in; (void)out_size; (void)d_ws; (void)ws_size;
    // inputs: 0=g, 1=Wq_w, 2=Wq_b, 3=Wk_w, 4=Wk_b, 5=Wv_w, 6=Wv_b (q/k are dead)
    const float* g  = (const float*)d_in[0];
    const float* Wv = (const float*)d_in[5];
    const float* bv = (const float*)d_in[6];
    float* out = (float*)d_out;
    dim3 grid(GN / TILE, GN / TILE);   // 48 x 48 (i,j) tiles
    axis_attn_fused_wmma<<<grid, dim3(256), 0, stream>>>(g, Wv, bv, out);
}